// MIND_2D_loss_69853348102680
// MI455X (gfx1250) — compile-verified
//
#include <hip/hip_runtime.h>

// ---------------------------------------------------------------------------
// MIND-SSC 2D loss for MI455X (gfx1250, wave32).
// Memory-bound (67MB in, ~3us HBM floor; fits in 192MB L2 so pass-2 re-reads
// are L2 hits). Two passes because the var-clamp bound is a *global* mean.
// 5x5 box filter = two banded-matrix matmuls on the matrix pipe via
// V_WMMA_F32_16X16X4_F32 (K=20 split into 5 slabs of K=4).
// Each wave owns one 16x16 tile + private LDS slice -> no block barriers.
// Band operands pinned in VGPRs (asm barrier), scaled by 0.2 so the two
// stages compose to the 1/25 box average for free.
// diff2 stored TRANSPOSED (col-major, even stride) so every WMMA B operand is
// one aligned ds_load_b64 into an even VGPR pair -> no assembly movs.
// ---------------------------------------------------------------------------

#define IMG_H 512
#define IMG_W 512
#define BATCH 32
// tiles per image: 32 (b) * 32 * 32 = 32768; 4 waves/block -> 8192 blocks
#define TILES_PER_IMG (BATCH * (IMG_H / 16) * (IMG_W / 16))
#define WAVES_PB 4
#define BLOCKS_P (TILES_PER_IMG / WAVES_PB)

// per-wave LDS slice (floats):
//   img   24x25                  = 600
//   diff2 4ch x (32 cols x 20)   = 2560   (transposed: dC[c][j*20+i])
//   tmpV  16x34                  = 544
#define IMG_S 600
#define CH_S  640                  // 32 cols * 20 rows, cols 20..31 zero pad
#define D_S   (4 * CH_S)
#define T_OFF (IMG_S + D_S)        // 3160 floats -> 8B aligned
#define SLICE 3712                 // 4 waves * 3712 * 4B = 59392 B static LDS

typedef __attribute__((ext_vector_type(2))) float v2f;
typedef __attribute__((ext_vector_type(8))) float v8f;

__device__ __forceinline__ v8f wmma_f32(v2f a, v2f b, v8f c) {
  // D(16x16) = A(16x4) x B(4x16) + C   (fp32 WMMA, 8-arg form)
  return __builtin_amdgcn_wmma_f32_16x16x4_f32(
      /*neg_a=*/false, a, /*neg_b=*/false, b,
      /*c_mod=*/(short)0, c, /*reuse_a=*/false, /*reuse_b=*/false);
}

__device__ __forceinline__ int clampi(int v, int lo, int hi) {
  return v < lo ? lo : (v > hi ? hi : v);
}

// Banded operand slabs: band[k] holds K=4k..4k+3 of the 16x20 matrix with
// 0.2 iff m <= k <= m+4 (else 0), in WMMA A layout — also valid as the
// stage-2 B operand (same condition). Two applications give the 1/25 average.
// asm barrier pins the values in VGPRs so they are computed exactly once.
__device__ __forceinline__ void make_band(int lane, v2f band[5]) {
  const int M = lane & 15;
  const int koff = (lane >> 4) << 1;  // lanes 16-31 hold K+2 in the slab
#pragma unroll
  for (int k = 0; k < 5; ++k) {
    int k0 = 4 * k + koff;
    float b0 = (k0 >= M && k0 <= M + 4) ? 0.2f : 0.0f;
    float b1 = (k0 + 1 >= M && k0 + 1 <= M + 4) ? 0.2f : 0.0f;
    asm("" : "+v"(b0), "+v"(b1));  // opaque: block rematerialization
    band[k][0] = b0;
    band[k][1] = b1;
  }
}

// Zero diff2 pad columns j in [20,32) of all 4 channels: in the transposed
// layout this is the contiguous index range [400,640) of each channel.
__device__ __forceinline__ void zero_pad(int lane, float* ldsD) {
  for (int idx = lane; idx < 4 * 240; idx += 32) {
    int c = idx / 240, r = idx - c * 240;
    ldsD[c * CH_S + 400 + r] = 0.0f;
  }
}

// Compute ssd[4] (post box-average) for one 16x16 tile; lane holds 8 pixels:
// pixel (row = v + (lane<16?0:8), col = lane&15) of the tile, per D layout.
__device__ __forceinline__ void compute_ssd_tile(
    const float* __restrict__ img, int ty0, int tx0, int lane,
    const v2f band[5], float* ldsImg, float* ldsD, float* ldsT, v8f ssd[4]) {
  // --- 24x24 edge-clamped image halo (stride 25). All replication padding in
  // the reference reduces to clamped indices here (clamp composes).
  for (int idx = lane; idx < 24 * 24; idx += 32) {
    int i = idx / 24, j = idx - i * 24;
    int gy = clampi(ty0 - 4 + i, 0, IMG_H - 1);
    int gx = clampi(tx0 - 4 + j, 0, IMG_W - 1);
    ldsImg[i * 25 + j] = img[gy * IMG_W + gx];
  }

  // --- diff2, all 4 channels fused, on the 20x20 (tile + radius-2) region,
  // stored transposed: dC[j*20 + i] = diff2[row i][col j]. Store address is
  // exactly idx -> consecutive lanes hit consecutive LDS dwords.
  // Channels share 4 shifted samples: L=S(0,-2) R=S(0,+2) U=S(-2,0) D=S(+2,0)
  // ch0=(L-U)^2 ch1=(R-U)^2 ch2=(D-L)^2 ch3=(D-R)^2.
  for (int idx = lane; idx < 400; idx += 32) {
    int j = idx / 20, i = idx - j * 20;
    int yb = clampi(ty0 - 2 + i, 0, IMG_H - 1) - (ty0 - 4);  // [2,21]
    int xb = clampi(tx0 - 2 + j, 0, IMG_W - 1) - (tx0 - 4);  // [2,21]
    const float* p = ldsImg + yb * 25 + xb;
    float sL = p[-2], sR = p[2], sU = p[-2 * 25], sD = p[2 * 25];
    float d0 = sL - sU, d1 = sR - sU, d2 = sD - sL, d3 = sD - sR;
    ldsD[idx] = d0 * d0;
    ldsD[CH_S + idx] = d1 * d1;
    ldsD[2 * CH_S + idx] = d2 * d2;
    ldsD[3 * CH_S + idx] = d3 * d3;
  }

  const int M = lane & 15;               // A-row / B-col / D-col lane index
  const int koff = (lane >> 4) << 1;     // K sub-offset: lanes 16-31 hold K+2
  const int rowOff = (lane < 16) ? 0 : 8;

#pragma unroll
  for (int c = 0; c < 4; ++c) {
    const float* dC = ldsD + c * CH_S;

    // --- stage 1: tmpV(16x20) = Pv(16x20) * diff2(20x20)
    // B operand {diff2[k0][n], diff2[k0+1][n]} = contiguous aligned pair in
    // the transposed layout (k0 even, col stride 20 even) -> ds_load_b64.
    v8f acc0 = {};  // tmpV cols 0..15
    v8f acc1 = {};  // tmpV cols 16..19 (cols 20..31 fed by zero pad)
#pragma unroll
    for (int k = 0; k < 5; ++k) {
      int k0 = 4 * k + koff;
      v2f b0 = *reinterpret_cast<const v2f*>(dC + M * 20 + k0);
      v2f b1 = *reinterpret_cast<const v2f*>(dC + (M + 16) * 20 + k0);
      acc0 = wmma_f32(band[k], b0, acc0);
      acc1 = wmma_f32(band[k], b1, acc1);
    }
    // bounce tmpV D-layout -> LDS so stage 2 can read it in A-layout
#pragma unroll
    for (int v = 0; v < 8; ++v) {
      ldsT[(v + rowOff) * 34 + M] = acc0[v];
      ldsT[(v + rowOff) * 34 + M + 16] = acc1[v];
    }

    // --- stage 2: ssd(16x16) = tmpV(16x20) * Qh(20x16); Qh slab == band slab
    // A operand {tmpV[m][k0], tmpV[m][k0+1]} contiguous aligned (stride 34).
    v8f acc2 = {};
#pragma unroll
    for (int k = 0; k < 5; ++k) {
      int k0 = 4 * k + koff;
      v2f a2 = *reinterpret_cast<const v2f*>(ldsT + M * 34 + k0);
      acc2 = wmma_f32(a2, band[k], acc2);
    }
    ssd[c] = acc2;  // 1/25 already folded into band*band
  }
}

__device__ __forceinline__ float wave_sum(float s) {
#pragma unroll
  for (int off = 16; off > 0; off >>= 1) s += __shfl_xor(s, off, 32);
  return s;
}

// ---- pass 1: global sum of mind_var for each image (ws[0]=pred, ws[1]=true)
__global__ void __launch_bounds__(32 * WAVES_PB) mind_pass1(
    const float* __restrict__ yp, const float* __restrict__ yt, float* ws) {
  __shared__ float lds[WAVES_PB * SLICE];
  int wave = threadIdx.x >> 5, lane = threadIdx.x & 31;
  int tile = blockIdx.x * WAVES_PB + wave;
  int b = tile >> 10, rem = tile & 1023;
  int ty0 = (rem >> 5) << 4, tx0 = (rem & 31) << 4;
  const float* img =
      (blockIdx.y == 0 ? yp : yt) + (size_t)b * (IMG_H * IMG_W);
  float* base = &lds[wave * SLICE];
  v2f band[5];
  make_band(lane, band);
  zero_pad(lane, base + IMG_S);
  v8f ssd[4];
  compute_ssd_tile(img, ty0, tx0, lane, band, base, base + IMG_S, base + T_OFF,
                   ssd);
  float s = 0.0f;
#pragma unroll
  for (int v = 0; v < 8; ++v) {
    float a0 = ssd[0][v], a1 = ssd[1][v], a2 = ssd[2][v], a3 = ssd[3][v];
    float mn = fminf(fminf(a0, a1), fminf(a2, a3));
    s += (a0 + a1 + a2 + a3) * 0.25f - mn;  // mean(mind) = mean(ssd) - min
  }
  s = wave_sum(s);
  if (lane == 0) atomicAdd(&ws[blockIdx.y], s);  // global_atomic_add_f32
}

// ---- pass 2: recompute both descriptors (L2-resident re-read), MSE-accumulate
__global__ void __launch_bounds__(32 * WAVES_PB) mind_pass2(
    const float* __restrict__ yp, const float* __restrict__ yt, float* ws) {
  __shared__ float lds[WAVES_PB * SLICE];
  int wave = threadIdx.x >> 5, lane = threadIdx.x & 31;
  int tile = blockIdx.x * WAVES_PB + wave;
  int b = tile >> 10, rem = tile & 1023;
  int ty0 = (rem >> 5) << 4, tx0 = (rem & 31) << 4;
  const float invN = 1.0f / (float)(BATCH * IMG_H * IMG_W);
  float mP = ws[0] * invN, mT = ws[1] * invN;  // global means from pass 1
  float* base = &lds[wave * SLICE];
  v2f band[5];
  make_band(lane, band);
  zero_pad(lane, base + IMG_S);
  v8f sp[4], st[4];
  compute_ssd_tile(yp + (size_t)b * (IMG_H * IMG_W), ty0, tx0, lane, band,
                   base, base + IMG_S, base + T_OFF, sp);
  compute_ssd_tile(yt + (size_t)b * (IMG_H * IMG_W), ty0, tx0, lane, band,
                   base, base + IMG_S, base + T_OFF, st);
  float s = 0.0f;
#pragma unroll
  for (int v = 0; v < 8; ++v) {
    float p0 = sp[0][v], p1 = sp[1][v], p2 = sp[2][v], p3 = sp[3][v];
    float t0 = st[0][v], t1 = st[1][v], t2 = st[2][v], t3 = st[3][v];
    float mnp = fminf(fminf(p0, p1), fminf(p2, p3));
    float mnt = fminf(fminf(t0, t1), fminf(t2, t3));
    float vp = (p0 + p1 + p2 + p3) * 0.25f - mnp;
    float vt = (t0 + t1 + t2 + t3) * 0.25f - mnt;
    vp = fminf(fmaxf(vp, mP * 0.001f), mP * 1000.0f);
    vt = fminf(fmaxf(vt, mT * 0.001f), mT * 1000.0f);
    float ivp = -1.0f / vp, ivt = -1.0f / vt;
    float d;
    d = __expf((p0 - mnp) * ivp) - __expf((t0 - mnt) * ivt); s += d * d;
    d = __expf((p1 - mnp) * ivp) - __expf((t1 - mnt) * ivt); s += d * d;
    d = __expf((p2 - mnp) * ivp) - __expf((t2 - mnt) * ivt); s += d * d;
    d = __expf((p3 - mnp) * ivp) - __expf((t3 - mnt) * ivt); s += d * d;
  }
  s = wave_sum(s);
  if (lane == 0) atomicAdd(&ws[2], s);
}

__global__ void mind_zero(float* ws) { ws[0] = 0.0f; ws[1] = 0.0f; ws[2] = 0.0f; }

__global__ void mind_fin(const float* __restrict__ ws, float* __restrict__ out) {
  out[0] = ws[2] * (1.0f / (float)(BATCH * 4 * IMG_H * IMG_W));
}

extern "C" void kernel_launch(void* const* d_in, const int* in_sizes, int n_in,
                              void* d_out, int out_size, void* d_ws,
                              size_t ws_size, hipStream_t stream) {
  const float* yp = (const float*)d_in[0];
  const float* yt = (const float*)d_in[1];
  float* ws = (float*)d_ws;   // ws[0]=sum var(pred), ws[1]=sum var(true), ws[2]=sum sqdiff
  float* out = (float*)d_out;

  mind_zero<<<1, 1, 0, stream>>>(ws);
  mind_pass1<<<dim3(BLOCKS_P, 2), 32 * WAVES_PB, 0, stream>>>(yp, yt, ws);
  mind_pass2<<<dim3(BLOCKS_P), 32 * WAVES_PB, 0, stream>>>(yp, yt, ws);
  mind_fin<<<1, 1, 0, stream>>>(ws, out);
}